// KeypointSpatialAttention_53635551592534
// MI455X (gfx1250) — compile-verified
//
#include <hip/hip_runtime.h>
#include <hip/hip_bf16.h>
#include <math.h>

// Problem constants (from reference)
#define NB   2048
#define CC   1024
#define FHW  49        // 7*7
#define NKP  6
#define HID  128
#define OUTD 256
#define MROWS (NB * NKP)          // 12288
#define MTILES (MROWS / 16)       // 768

typedef __attribute__((ext_vector_type(2))) float v2f;
typedef __attribute__((ext_vector_type(8))) float v8f;

// ---------------------------------------------------------------------------
// Kernel 1: fused bilinear gather + GEMM1 (S @ W1) + bias + exact GELU
//   block: 256 threads (8 wave32), owns 16 rows of S (rows = n*6+k)
//   LDS:   As[16][260] fp32 chunk of the sampled matrix (K-chunks of 256)
//   Gather params pre-combined as int32 element index (n*C*49 + y*7 + x)
//   so corner loads use SADDR + 32-bit index + scale_offset addressing.
// ---------------------------------------------------------------------------
__global__ void __launch_bounds__(256)
kp_gather_gemm1(const float* __restrict__ feat,   // (N,C,7,7)
                const float* __restrict__ kp,     // (N,6,2)
                const float* __restrict__ W1,     // (1024,128) row-major
                const float* __restrict__ b1,     // (128)
                const int* __restrict__ cox, const int* __restrict__ coy,
                const int* __restrict__ cw,  const int* __restrict__ ch,
                const int* __restrict__ iw,  const int* __restrict__ ih,
                float* __restrict__ h_out,        // (12288,128)
                float* __restrict__ validf)       // (12288)
{
    __shared__ float As[16][260];      // 256-wide K chunk, +4 pad (bank skew)
    __shared__ float4 wgt4[16];        // 4 bilinear weights (valid folded in)
    __shared__ int4   idx4[16];        // 4 combined element indices (signed)

    const int tid  = threadIdx.x;
    const int row0 = blockIdx.x * 16;

    if (tid < 16) {
        const int r = row0 + tid;
        const int n = r / NKP;
        const float u = kp[(size_t)r * 2 + 0];
        const float v = kp[(size_t)r * 2 + 1];
        const float pxx = u * (float)(*iw);
        const float pxy = v * (float)(*ih);
        const float cx  = (pxx - (float)(*cox)) / (float)(*cw);
        const float cy  = (pxy - (float)(*coy)) / (float)(*ch);
        const bool invalid = (u < 0.f) || (v < 0.f) ||
                             (cx < 0.f) || (cx > 1.f) ||
                             (cy < 0.f) || (cy > 1.f);
        const float val = invalid ? 0.f : 1.f;
        // align_corners=True: ix = crop_x * (W-1)
        const float ixx = cx * 6.f;
        const float iyy = cy * 6.f;
        const float x0 = floorf(ixx), y0 = floorf(iyy);
        const float wx1 = ixx - x0, wx0 = 1.f - wx1;
        const float wy1 = iyy - y0, wy0 = 1.f - wy1;
        const float xs[2]  = {x0, x0 + 1.f};
        const float ys[2]  = {y0, y0 + 1.f};
        const float wxs[2] = {wx0, wx1};
        const float wys[2] = {wy0, wy1};
        const int nbase = n * (CC * FHW);            // <= 102.7M, fits int32
        float w[4];
        int   e[4];
#pragma unroll
        for (int jy = 0; jy < 2; ++jy)
#pragma unroll
            for (int jx = 0; jx < 2; ++jx) {
                const float xi = xs[jx], yi = ys[jy];
                const bool inb = (xi >= 0.f) && (xi <= 6.f) &&
                                 (yi >= 0.f) && (yi <= 6.f);
                const int xc = (int)fminf(fmaxf(xi, 0.f), 6.f);
                const int yc = (int)fminf(fmaxf(yi, 0.f), 6.f);
                e[jy * 2 + jx] = nbase + yc * 7 + xc;
                w[jy * 2 + jx] = wxs[jx] * wys[jy] * (inb ? 1.f : 0.f) * val;
            }
        wgt4[tid] = make_float4(w[0], w[1], w[2], w[3]);
        idx4[tid] = make_int4(e[0], e[1], e[2], e[3]);
        validf[r] = val;
    }
    __syncthreads();

    const int lane = tid & 31;
    const int wave = tid >> 5;
    const int hi   = lane >> 4;        // K sub-group (ISA 32-bit A/B layout)
    const int m    = lane & 15;
    const int col  = wave * 16 + m;    // output column in [0,128)

    v8f acc = {};

    for (int kc = 0; kc < 4; ++kc) {
        // ---- gather 16 rows x 256 channels of S into LDS ----
        const int c     = kc * 256 + tid;     // channel handled by me
        const int cbase = c * FHW;            // element offset of my channel
#pragma unroll 4
        for (int r = 0; r < 16; ++r) {
            const int4   e = idx4[r];                 // ds_load_b128
            const float4 w = wgt4[r];                 // ds_load_b128
            // signed adds: compiler folds sext(cbase+e) -> saddr+scale_offset
            const float s = w.x * feat[cbase + e.x] + w.y * feat[cbase + e.y]
                          + w.z * feat[cbase + e.z] + w.w * feat[cbase + e.w];
            As[r][c - kc * 256] = s;
        }
        __syncthreads();

        // ---- 64 wmma f32 16x16x4 steps over this chunk ----
#pragma unroll 4
        for (int kk = 0; kk < 256; kk += 4) {
            const int ks = kk + 2 * hi;
            v2f a = *(const v2f*)&As[m][ks];
            const int kg = kc * 256 + ks;
            v2f b;
            b.x = W1[(size_t)kg * HID + col];
            b.y = W1[(size_t)(kg + 1) * HID + col];
            acc = __builtin_amdgcn_wmma_f32_16x16x4_f32(
                false, a, false, b, (short)0, acc, false, false);
        }
        __syncthreads();
    }

    // epilogue: + b1, exact GELU, store h
    const float bb = b1[col];
#pragma unroll
    for (int r = 0; r < 8; ++r) {
        const int mr = r + 8 * hi;
        const float x = acc[r] + bb;
        const float g = 0.5f * x * (1.f + erff(x * 0.70710678118654752f));
        h_out[(size_t)(row0 + mr) * HID + col] = g;
    }
}

// ---------------------------------------------------------------------------
// Kernel 2: GEMM2 (h @ W2 + b2) -> kpf (12288 x 256)
//   grid (768, 2); block 256 threads; each block: 16 rows x 128 cols
// ---------------------------------------------------------------------------
__global__ void __launch_bounds__(256)
kp_gemm2(const float* __restrict__ h,      // (12288,128)
         const float* __restrict__ W2,     // (128,256) row-major
         const float* __restrict__ b2,     // (256)
         float* __restrict__ kpf)          // (12288,256)
{
    __shared__ float As[16][132];          // 16 x 128, +4 pad

    const int tid  = threadIdx.x;
    const int row0 = blockIdx.x * 16;
    const int cb   = blockIdx.y * 128;

#pragma unroll
    for (int i = tid; i < 16 * 128; i += 256) {
        const int r = i >> 7, cc = i & 127;
        As[r][cc] = h[(size_t)(row0 + r) * HID + cc];
    }
    __syncthreads();

    const int lane = tid & 31;
    const int wave = tid >> 5;
    const int hi   = lane >> 4;
    const int m    = lane & 15;
    const int col  = cb + wave * 16 + m;

    v8f acc = {};
#pragma unroll 4
    for (int kk = 0; kk < 128; kk += 4) {
        const int ks = kk + 2 * hi;
        v2f a = *(const v2f*)&As[m][ks];
        v2f b;
        b.x = W2[(size_t)ks * OUTD + col];
        b.y = W2[(size_t)(ks + 1) * OUTD + col];
        acc = __builtin_amdgcn_wmma_f32_16x16x4_f32(
            false, a, false, b, (short)0, acc, false, false);
    }

    const float bb = b2[col];
#pragma unroll
    for (int r = 0; r < 8; ++r) {
        const int mr = r + 8 * hi;
        kpf[(size_t)(row0 + mr) * OUTD + col] = acc[r] + bb;
    }
}

// ---------------------------------------------------------------------------
// Kernel 3: per-sample reduction over the 6 keypoints, divide by n_valid
// ---------------------------------------------------------------------------
__global__ void __launch_bounds__(256)
kp_reduce(const float* __restrict__ kpf,    // (12288,256)
          const float* __restrict__ validf, // (12288)
          float* __restrict__ out)          // (2048,256)
{
    const int idx = blockIdx.x * 256 + threadIdx.x;   // n*256 + o
    const int n = idx >> 8;
    const int o = idx & 255;

    float s = 0.f;
#pragma unroll
    for (int k = 0; k < NKP; ++k)
        s += kpf[(size_t)(n * NKP + k) * OUTD + o];

    float nv = 0.f;
#pragma unroll
    for (int k = 0; k < NKP; ++k)
        nv += validf[n * NKP + k];
    nv = fmaxf(nv, 1.f);

    out[idx] = s / nv;
}

// ---------------------------------------------------------------------------
extern "C" void kernel_launch(void* const* d_in, const int* in_sizes, int n_in,
                              void* d_out, int out_size, void* d_ws, size_t ws_size,
                              hipStream_t stream) {
    (void)in_sizes; (void)n_in; (void)out_size; (void)ws_size;

    const float* feat = (const float*)d_in[0];
    const float* kp   = (const float*)d_in[1];
    const float* W1   = (const float*)d_in[2];
    const float* b1   = (const float*)d_in[3];
    const float* W2   = (const float*)d_in[4];
    const float* b2   = (const float*)d_in[5];
    const int* cox = (const int*)d_in[6];
    const int* coy = (const int*)d_in[7];
    const int* cw  = (const int*)d_in[8];
    const int* ch  = (const int*)d_in[9];
    const int* iw  = (const int*)d_in[10];
    const int* ih  = (const int*)d_in[11];
    float* out = (float*)d_out;

    // workspace layout (fp32): validf[12288] | h[12288*128] | kpf[12288*256]
    float* validf = (float*)d_ws;
    float* h      = validf + MROWS;
    float* kpf    = h + (size_t)MROWS * HID;

    kp_gather_gemm1<<<MTILES, 256, 0, stream>>>(feat, kp, W1, b1,
                                                cox, coy, cw, ch, iw, ih,
                                                h, validf);
    kp_gemm2<<<dim3(MTILES, 2), 256, 0, stream>>>(h, W2, b2, kpf);
    kp_reduce<<<(NB * OUTD) / 256, 256, 0, stream>>>(kpf, validf, out);
}